// set_loss_54537494724893
// MI455X (gfx1250) — compile-verified
//
#include <hip/hip_runtime.h>

typedef __attribute__((ext_vector_type(2))) float v2f;
typedef __attribute__((ext_vector_type(8))) float v8f;

#define NPTS 16384

// Each wave owns one 16-row tile of loc1 and sweeps all 1024 16-col tiles of
// loc2.  dot products go through V_WMMA_F32_16X16X4_F32 (K padded 2 -> 4 with
// zeros in lanes 16-31 of the A/B fragments); the row-min is fused in
// registers:  min_n(sq1+sq2-2d) = sq1 + min_n(sq2-2d).
__global__ __launch_bounds__(256) void nn_rowmin_kernel(
    const float* __restrict__ loc1, const float* __restrict__ loc2,
    float* __restrict__ rowmin) {
  const int lane = threadIdx.x & 31;
  const int wave = threadIdx.x >> 5;
  const int m0   = (blockIdx.x * 8 + wave) * 16;
  const int l15  = lane & 15;
  const bool lo  = lane < 16;

  // A fragment: lanes 0-15 = (x,y) of row m0+lane (K=0,1); lanes 16-31 = K=2,3 pad = 0.
  const float2 p1 = *(const float2*)(loc1 + (size_t)(m0 + l15) * 2);
  v2f a;
  a.x = lo ? p1.x : 0.0f;
  a.y = lo ? p1.y : 0.0f;
  const float sq1 = p1.x * p1.x + p1.y * p1.y;  // ||loc1[m0+l15]||^2

  float mn[8];
#pragma unroll
  for (int i = 0; i < 8; ++i) mn[i] = 3.4e38f;

  for (int n0 = 0; n0 < NPTS; n0 += 16) {
    // All lanes load the column point for N = n0 + (lane&15); lanes 16-31
    // reuse it only for sq2 (their B slots are the K=2,3 zero pad).
    const float2 p2 = *(const float2*)(loc2 + (size_t)(n0 + l15) * 2);
    const float s2  = p2.x * p2.x + p2.y * p2.y;
    v2f b;
    b.x = lo ? p2.x : 0.0f;
    b.y = lo ? p2.y : 0.0f;

    v8f d = {};  // C = 0
    d = __builtin_amdgcn_wmma_f32_16x16x4_f32(
        /*neg_a=*/false, a, /*neg_b=*/false, b,
        /*c_mod=*/(short)0, d, /*reuse_a=*/false, /*reuse_b=*/false);

    // D layout: lane L, vgpr i  ->  (M = i + 8*(L>=16), N = L&15)
#pragma unroll
    for (int i = 0; i < 8; ++i) {
      const float g = fmaf(-2.0f, d[i], s2);  // sq2[n] - 2*dot
      mn[i] = fminf(mn[i], g);
    }
  }

  // Min-reduce across the 16 lanes of each half (same M, different N columns).
  // xor offsets < 16 never cross the 16-lane halves on wave32.
#pragma unroll
  for (int off = 1; off < 16; off <<= 1) {
#pragma unroll
    for (int i = 0; i < 8; ++i) {
      mn[i] = fminf(mn[i], __shfl_xor(mn[i], off, 32));
    }
  }

  // Writer lanes: v in 0..7 writes row m0+v (half 0), lane 16+v writes row
  // m0+8+v (half 1).  Fetch the matching ||loc1[m]||^2 via shuffle:
  //   lane v    <- lane v     (sq1 of m0+v)
  //   lane 16+v <- lane 8+v   (sq1 of m0+8+v)
  const float sq1w = __shfl(sq1, lo ? lane : (lane - 8), 32);
  if (l15 < 8) {
    const int half = lane >> 4;
    const int v    = lane & 7;
    rowmin[m0 + half * 8 + v] = sq1w + mn[v];
  }
}

// Deterministic single-block mean of the 16384 row-mins (fixed-order strided
// accumulation + fixed LDS tree; no float atomics -> bit-stable across replays).
__global__ __launch_bounds__(256) void mean_kernel(const float* __restrict__ rowmin,
                                                   float* __restrict__ out) {
  __shared__ float s[256];
  float acc = 0.0f;
  for (int i = threadIdx.x; i < NPTS; i += 256) acc += rowmin[i];
  s[threadIdx.x] = acc;
  __syncthreads();
#pragma unroll
  for (int off = 128; off > 0; off >>= 1) {
    if ((int)threadIdx.x < off) s[threadIdx.x] += s[threadIdx.x + off];
    __syncthreads();
  }
  if (threadIdx.x == 0) out[0] = s[0] * (1.0f / (float)NPTS);
}

extern "C" void kernel_launch(void* const* d_in, const int* in_sizes, int n_in,
                              void* d_out, int out_size, void* d_ws, size_t ws_size,
                              hipStream_t stream) {
  const float* loc1 = (const float*)d_in[0];
  const float* loc2 = (const float*)d_in[1];
  float* rowmin = (float*)d_ws;  // 16384 floats = 64 KB of scratch

  // 128 blocks x 8 waves x 16 rows = 16384 rows
  nn_rowmin_kernel<<<NPTS / 128, 256, 0, stream>>>(loc1, loc2, rowmin);
  mean_kernel<<<1, 256, 0, stream>>>(rowmin, (float*)d_out);
}